// MoERankingModel_42743514530370
// MI455X (gfx1250) — compile-verified
//
#include <hip/hip_runtime.h>
#include <hip/hip_bf16.h>

typedef __attribute__((ext_vector_type(2))) float v2f;
typedef __attribute__((ext_vector_type(4))) float v4f;
typedef __attribute__((ext_vector_type(8))) float v8f;
typedef __attribute__((ext_vector_type(2))) int   v2i;
typedef __attribute__((ext_vector_type(4))) int   v4i;

#define WMMA_F32X4(a, b, c) \
  __builtin_amdgcn_wmma_f32_16x16x4_f32(false, (a), false, (b), (short)0, (c), false, false)

static constexpr int BATCH = 131072;
static constexpr int DIN   = 512;
static constexpr int HD    = 256;
static constexpr int NE    = 8;
static constexpr int BKP   = 20;   // A pitch: 16B-aligned rows, conflict-free b64 frag reads
static constexpr int CAP   = 2 * BATCH + NE * 32;

// -------- async global->LDS path (gfx1250 ASYNCcnt), with synchronous fallback ----------

#ifndef __has_builtin
#define __has_builtin(x) 0
#endif
#if __has_builtin(__builtin_amdgcn_global_load_async_to_lds_b128) && \
    __has_builtin(__builtin_amdgcn_global_load_async_to_lds_b64)  && \
    __has_builtin(__builtin_amdgcn_s_wait_asynccnt)
#define ASYNC_LDS 1
#else
#define ASYNC_LDS 0
#endif

typedef __attribute__((address_space(1))) v4i gv4i;   // global-AS int4
typedef __attribute__((address_space(3))) v4i lv4i;   // LDS-AS int4
typedef __attribute__((address_space(1))) v2i gv2i;   // global-AS int2
typedef __attribute__((address_space(3))) v2i lv2i;   // LDS-AS int2

__device__ __forceinline__ void lds_cp_b128(float* l, const float* g) {
#if ASYNC_LDS
  __builtin_amdgcn_global_load_async_to_lds_b128((gv4i*)g, (lv4i*)l, 0, 0);
#else
  *(v4f*)l = *(const v4f*)g;
#endif
}
__device__ __forceinline__ void lds_cp_b64(float* l, const float* g) {
#if ASYNC_LDS
  __builtin_amdgcn_global_load_async_to_lds_b64((gv2i*)g, (lv2i*)l, 0, 0);
#else
  *(v2f*)l = *(const v2f*)g;
#endif
}
__device__ __forceinline__ void async_wait() {
#if ASYNC_LDS
  __builtin_amdgcn_s_wait_asynccnt(0);
#endif
}

// -------- staging issue helpers (256 threads, uniform per wave) -------------------------

// 64 x 16 A tile: one b128 per thread.
__device__ __forceinline__ void issue_A64(float* As, const float* __restrict__ A, int lda,
                                          int row0, int k0, int tid) {
  int r = tid >> 2, c4 = (tid & 3) << 2;
  lds_cp_b128(As + r * BKP + c4, A + (size_t)(row0 + r) * lda + k0 + c4);
}

// 32 x 16 gathered A tile: one b64 per thread (row indices from LDS).
__device__ __forceinline__ void issue_A32g(float* As, const float* __restrict__ A, int lda,
                                           const int* stok, int k0, int tid) {
  int r = tid >> 3, c2 = (tid & 7) << 1;
  int g = stok[r];
  size_t rr = (size_t)(g < 0 ? 0 : g);
  lds_cp_b64(As + r * BKP + c2, A + rr * lda + k0 + c2);
}

// 16 x BN B tile (row stride == BN): b128s, BN/64 per thread.
template <int BN>
__device__ __forceinline__ void issue_B(float* Bs, const float* __restrict__ Bsrc, int k0, int tid) {
#pragma unroll
  for (int f = tid; f < BN * 4; f += 256) {
    int r = f / (BN / 4), c4 = (f % (BN / 4)) << 2;
    lds_cp_b128(Bs + r * (BN + 4) + c4, Bsrc + (size_t)(k0 + r) * BN + c4);
  }
}

// One K=16 step: 4 chained 16x16x4 fp32 WMMAs over NT 16-col tiles (ISA 7.12.2 layouts).
template <int NT>
__device__ __forceinline__ void wmma_k16(const float* ArowK, const float* Bs, int bnp,
                                         int colbase, int hi, v8f* acc) {
#pragma unroll
  for (int kk = 0; kk < 4; ++kk) {
    const int ks = kk * 4 + 2 * hi;
    v2f a = *(const v2f*)(ArowK + ks);
#pragma unroll
    for (int ct = 0; ct < NT; ++ct) {
      const float* bp = Bs + ks * bnp + colbase + ct * 16;
      v2f b; b.x = bp[0]; b.y = bp[bnp];
      acc[ct] = WMMA_F32X4(a, b, acc[ct]);
    }
  }
}

// ---------------- K0: per-call workspace init -------------------------------------------

__global__ __launch_bounds__(256) void k_init(float* moe, int* slot_token, int* cnt, int* cur) {
  unsigned i = blockIdx.x * 256u + threadIdx.x;
  if (i < (unsigned)(BATCH * HD)) moe[i] = 0.0f;
  if (i < (unsigned)CAP) slot_token[i] = -1;
  if (i < NE) { cnt[i] = 0; cur[i] = 0; }
}

// ---------------- Wf = W_in @ Wg  ([512,8], tiny) ---------------------------------------

__global__ __launch_bounds__(256) void k_fuse_gate(const float* __restrict__ Win,
                                                   const float* __restrict__ Wg,
                                                   float* __restrict__ Wf) {
  int i = blockIdx.x * 256 + threadIdx.x;
  if (i >= DIN * NE) return;
  int k = i >> 3, e = i & 7;
  float s = 0.0f;
  for (int m = 0; m < HD; ++m) s += Win[(size_t)k * HD + m] * Wg[(size_t)m * NE + e];
  Wf[i] = s;
}

// ---------------- K1: h = x@W_in + b_in, fused gating (top-2 + counts) ------------------
// 64x256 tile / block, 8 waves, double-buffered async pipeline.

__global__ __launch_bounds__(256) void k_in_gemm_gate(
    const float* __restrict__ x, const float* __restrict__ Win, const float* __restrict__ bin,
    const float* __restrict__ Wf, const float* __restrict__ bg,
    float* __restrict__ h, int* __restrict__ tope, float* __restrict__ topg,
    int* __restrict__ cnt) {
  __shared__ float As[2][64 * BKP];
  __shared__ float Bs[2][16 * (HD + 4)];
  __shared__ float sWf[DIN * NE];
  __shared__ float lgt[64 * NE];

  const int tid = threadIdx.x;
  const int row0 = blockIdx.x * 64;
  for (int i = tid; i < DIN * NE; i += 256) sWf[i] = Wf[i];

  const int wave = tid >> 5, lane = tid & 31, hi = lane >> 4, m = lane & 15;
  const int rt = wave & 3, half = (wave >> 2) * 128;
  const int gj = tid & 63, ge = (tid >> 6) * 2;

  v8f acc[8];
#pragma unroll
  for (int ct = 0; ct < 8; ++ct)
#pragma unroll
    for (int j = 0; j < 8; ++j) acc[ct][j] = 0.0f;

  issue_A64(As[0], x, DIN, row0, 0, tid);
  issue_B<HD>(Bs[0], Win, 0, tid);

  float g0 = 0.0f, g1 = 0.0f;
  for (int k0 = 0; k0 < DIN; k0 += 16) {
    const int cur = (k0 >> 4) & 1;
    async_wait();
    __syncthreads();
    if (k0 + 16 < DIN) {
      issue_A64(As[cur ^ 1], x, DIN, row0, k0 + 16, tid);
      issue_B<HD>(Bs[cur ^ 1], Win, k0 + 16, tid);
    }
    wmma_k16<8>(As[cur] + (rt * 16 + m) * BKP, Bs[cur], HD + 4, half + m, hi, acc);
    // fused gating partial: logits += x_tile @ Wf_tile
    const float* xr = As[cur] + gj * BKP;
    const float* wf = sWf + k0 * NE + ge;
#pragma unroll
    for (int k = 0; k < 16; ++k) {
      float xv = xr[k];
      g0 += xv * wf[k * NE];
      g1 += xv * wf[k * NE + 1];
    }
  }

  lgt[gj * NE + ge] = g0;
  lgt[gj * NE + ge + 1] = g1;

#pragma unroll
  for (int ct = 0; ct < 8; ++ct) {
    int col = half + ct * 16 + m;
    float bv = bin[col];
#pragma unroll
    for (int j = 0; j < 8; ++j) {
      int r = rt * 16 + j + hi * 8;
      h[(size_t)(row0 + r) * HD + col] = acc[ct][j] + bv;
    }
  }
  __syncthreads();

  if (tid < 64) {
    int t = row0 + tid;
    float l[NE];
#pragma unroll
    for (int e = 0; e < NE; ++e) l[e] = lgt[tid * NE + e] + bg[e];
    int e0 = 0; float v0 = l[0];
#pragma unroll
    for (int e = 1; e < NE; ++e) if (l[e] > v0) { v0 = l[e]; e0 = e; }
    int e1 = (e0 == 0) ? 1 : 0; float v1 = l[e1];
#pragma unroll
    for (int e = 0; e < NE; ++e) if (e != e0 && l[e] > v1) { v1 = l[e]; e1 = e; }
    float b = __expf(v1 - v0);
    float inv = 1.0f / (1.0f + b);
    tope[2 * t] = e0;  tope[2 * t + 1] = e1;
    topg[2 * t] = inv; topg[2 * t + 1] = b * inv;
    atomicAdd(&cnt[e0], 1);
    atomicAdd(&cnt[e1], 1);
  }
}

// ---------------- K2: padded (to 32) exclusive scan of per-expert counts -----------------

__global__ void k_scan(const int* __restrict__ cnt, int* __restrict__ padOff) {
  if (threadIdx.x == 0) {
    int off = 0;
    for (int e = 0; e < NE; ++e) { padOff[e] = off; off += ((cnt[e] + 31) >> 5) << 5; }
    padOff[NE] = off;
  }
}

// ---------------- K3: scatter tokens into per-expert slot segments -----------------------

__global__ __launch_bounds__(256) void k_scatter(const int* __restrict__ tope,
                                                 const float* __restrict__ topg,
                                                 const int* __restrict__ padOff,
                                                 int* __restrict__ cur,
                                                 int* __restrict__ slot_token,
                                                 float* __restrict__ slot_gate) {
  int t = blockIdx.x * 256 + threadIdx.x;
  if (t >= BATCH) return;
#pragma unroll
  for (int k = 0; k < 2; ++k) {
    int e = tope[2 * t + k];
    int pos = padOff[e] + atomicAdd(&cur[e], 1);
    slot_token[pos] = t;
    slot_gate[pos] = topg[2 * t + k];
  }
}

// ---------------- K4: grouped expert GEMM (gather -> GEMM1+relu -> GEMM2 -> scatter) -----
// 32x256 tile / block, 8 waves: rt = wave&1, half = (wave>>1)*64, NT=4.

__global__ __launch_bounds__(256) void k_expert(
    const float* __restrict__ h, const float* __restrict__ W1, const float* __restrict__ b1,
    const float* __restrict__ W2, const float* __restrict__ b2,
    const int* __restrict__ padOff, const int* __restrict__ slot_token,
    const float* __restrict__ slot_gate, float* __restrict__ moe) {
  __shared__ float As[2][32 * BKP];
  __shared__ float Bs[2][16 * (HD + 4)];
  __shared__ float E1[32 * (HD + 2)];
  __shared__ int   sOff[NE + 1];
  __shared__ int   stok[32];
  __shared__ float sg[32];

  const int tid = threadIdx.x;
  const int row0 = blockIdx.x * 32;
  if (tid < NE + 1) sOff[tid] = padOff[tid];
  __syncthreads();
  if (row0 >= sOff[NE]) return;   // uniform exit

  int e = 0;
  while (e < NE - 1 && row0 >= sOff[e + 1]) ++e;
  if (tid < 32) { stok[tid] = slot_token[row0 + tid]; sg[tid] = slot_gate[row0 + tid]; }
  __syncthreads();                // stok visible to all stagers

  const int wave = tid >> 5, lane = tid & 31, hi = lane >> 4, m = lane & 15;
  const int rt = wave & 1, half = (wave >> 1) * 64;
  const int E1P = HD + 2;

  v8f acc[4];
#pragma unroll
  for (int ct = 0; ct < 4; ++ct)
#pragma unroll
    for (int j = 0; j < 8; ++j) acc[ct][j] = 0.0f;

  // GEMM1: e1 = relu(h_gathered @ W1[e] + b1[e])
  const float* B1p = W1 + (size_t)e * HD * HD;
  issue_A32g(As[0], h, HD, stok, 0, tid);
  issue_B<HD>(Bs[0], B1p, 0, tid);
  for (int k0 = 0; k0 < HD; k0 += 16) {
    const int cur = (k0 >> 4) & 1;
    async_wait();
    __syncthreads();
    if (k0 + 16 < HD) {
      issue_A32g(As[cur ^ 1], h, HD, stok, k0 + 16, tid);
      issue_B<HD>(Bs[cur ^ 1], B1p, k0 + 16, tid);
    }
    wmma_k16<4>(As[cur] + (rt * 16 + m) * BKP, Bs[cur], HD + 4, half + m, hi, acc);
  }
  const float* b1p = b1 + (size_t)e * HD;
#pragma unroll
  for (int ct = 0; ct < 4; ++ct) {
    int col = half + ct * 16 + m;
    float bv = b1p[col];
#pragma unroll
    for (int j = 0; j < 8; ++j) {
      int r = rt * 16 + j + hi * 8;
      float v = acc[ct][j] + bv;
      E1[r * E1P + col] = v > 0.0f ? v : 0.0f;
      acc[ct][j] = 0.0f;
    }
  }

  // GEMM2: eo = e1 @ W2[e] + b2[e]; scale by gate; atomic-add into moe
  const float* B2p = W2 + (size_t)e * HD * HD;
  issue_B<HD>(Bs[0], B2p, 0, tid);
  for (int k0 = 0; k0 < HD; k0 += 16) {
    const int cur = (k0 >> 4) & 1;
    async_wait();
    __syncthreads();    // also makes all E1 writes visible on first iteration
    if (k0 + 16 < HD) issue_B<HD>(Bs[cur ^ 1], B2p, k0 + 16, tid);
    wmma_k16<4>(E1 + (rt * 16 + m) * E1P + k0, Bs[cur], HD + 4, half + m, hi, acc);
  }
  const float* b2p = b2 + (size_t)e * HD;
#pragma unroll
  for (int ct = 0; ct < 4; ++ct) {
    int col = half + ct * 16 + m;
    float bv = b2p[col];
#pragma unroll
    for (int j = 0; j < 8; ++j) {
      int r = rt * 16 + j + hi * 8;
      int tok = stok[r];
      if (tok >= 0) {
        float v = (acc[ct][j] + bv) * sg[r];
        unsafeAtomicAdd(&moe[(size_t)tok * HD + col], v);  // global_atomic_add_f32
      }
    }
  }
}

// ---------------- K5: head  y=relu(moe@Wh1+bh1); out = y@Wh2 + bh2 -----------------------
// 64x128 tile / block, 8 waves: rt = wave&3, half = (wave>>2)*64, NT=4.

__global__ __launch_bounds__(256) void k_head(
    const float* __restrict__ moe, const float* __restrict__ Wh1, const float* __restrict__ bh1,
    const float* __restrict__ Wh2, const float* __restrict__ bh2, float* __restrict__ out) {
  __shared__ float As[2][64 * BKP];
  __shared__ float Bs[2][16 * (128 + 4)];
  __shared__ float Y[64 * (128 + 4)];
  __shared__ float sW2[128];

  const int tid = threadIdx.x;
  const int row0 = blockIdx.x * 64;
  if (tid < 128) sW2[tid] = Wh2[tid];

  const int wave = tid >> 5, lane = tid & 31, hi = lane >> 4, m = lane & 15;
  const int rt = wave & 3, half = (wave >> 2) * 64;
  const int YP = 128 + 4;

  v8f acc[4];
#pragma unroll
  for (int ct = 0; ct < 4; ++ct)
#pragma unroll
    for (int j = 0; j < 8; ++j) acc[ct][j] = 0.0f;

  issue_A64(As[0], moe, HD, row0, 0, tid);
  issue_B<128>(Bs[0], Wh1, 0, tid);
  for (int k0 = 0; k0 < HD; k0 += 16) {
    const int cur = (k0 >> 4) & 1;
    async_wait();
    __syncthreads();
    if (k0 + 16 < HD) {
      issue_A64(As[cur ^ 1], moe, HD, row0, k0 + 16, tid);
      issue_B<128>(Bs[cur ^ 1], Wh1, k0 + 16, tid);
    }
    wmma_k16<4>(As[cur] + (rt * 16 + m) * BKP, Bs[cur], 128 + 4, half + m, hi, acc);
  }
#pragma unroll
  for (int ct = 0; ct < 4; ++ct) {
    int col = half + ct * 16 + m;
    float bv = bh1[col];
#pragma unroll
    for (int j = 0; j < 8; ++j) {
      int r = rt * 16 + j + hi * 8;
      float v = acc[ct][j] + bv;
      Y[r * YP + col] = v > 0.0f ? v : 0.0f;
    }
  }
  __syncthreads();
  if (tid < 64) {
    float s = bh2[0];
    const float* yr = Y + tid * YP;
#pragma unroll 8
    for (int c = 0; c < 128; ++c) s += yr[c] * sW2[c];
    out[row0 + tid] = s;
  }
}

// ---------------- host launcher ----------------------------------------------------------

extern "C" void kernel_launch(void* const* d_in, const int* in_sizes, int n_in,
                              void* d_out, int out_size, void* d_ws, size_t ws_size,
                              hipStream_t stream) {
  const float* x   = (const float*)d_in[0];
  const float* Win = (const float*)d_in[1];
  const float* bin = (const float*)d_in[2];
  const float* W1  = (const float*)d_in[3];
  const float* b1  = (const float*)d_in[4];
  const float* W2  = (const float*)d_in[5];
  const float* b2  = (const float*)d_in[6];
  const float* Wg  = (const float*)d_in[7];
  const float* bg  = (const float*)d_in[8];
  const float* Wh1 = (const float*)d_in[9];
  const float* bh1 = (const float*)d_in[10];
  const float* Wh2 = (const float*)d_in[11];
  const float* bh2 = (const float*)d_in[12];
  float* out = (float*)d_out;

  char* p = (char*)d_ws;
  float* h         = (float*)p; p += (size_t)BATCH * HD * 4;
  float* moe       = (float*)p; p += (size_t)BATCH * HD * 4;
  int*   tope      = (int*)p;   p += (size_t)BATCH * 2 * 4;
  float* topg      = (float*)p; p += (size_t)BATCH * 2 * 4;
  int*   slot_tok  = (int*)p;   p += (size_t)CAP * 4;
  float* slot_gate = (float*)p; p += (size_t)CAP * 4;
  float* Wf        = (float*)p; p += (size_t)DIN * NE * 4;
  int*   cnt       = (int*)p;   p += 64;
  int*   cur       = (int*)p;   p += 64;
  int*   padOff    = (int*)p;   p += 64;

  k_init<<<(BATCH * HD) / 256, 256, 0, stream>>>(moe, slot_tok, cnt, cur);
  k_fuse_gate<<<(DIN * NE + 255) / 256, 256, 0, stream>>>(Win, Wg, Wf);
  k_in_gemm_gate<<<BATCH / 64, 256, 0, stream>>>(x, Win, bin, Wf, bg, h, tope, topg, cnt);
  k_scan<<<1, 32, 0, stream>>>(cnt, padOff);
  k_scatter<<<BATCH / 256, 256, 0, stream>>>(tope, topg, padOff, cur, slot_tok, slot_gate);
  k_expert<<<CAP / 32, 256, 0, stream>>>(h, W1, b1, W2, b2, padOff, slot_tok, slot_gate, moe);
  k_head<<<BATCH / 64, 256, 0, stream>>>(moe, Wh1, bh1, Wh2, bh2, out);
}